// RiemannianGNN_47777216201088
// MI455X (gfx1250) — compile-verified
//
#include <hip/hip_runtime.h>
#include <hip/hip_bf16.h>
#include <math.h>

#define DD 128      // feature dim
#define KNB 32      // neighbors per node
#define EPSF 1e-5f
#define MAXTANH 15.0f

typedef __attribute__((ext_vector_type(2))) float v2f;
typedef __attribute__((ext_vector_type(8))) float v8f;

// ---------------------------------------------------------------------------
// prep: Y[n,:] = (do_log ? log_map_zero(X[n,:]) : X[n,:]) * mask[n]
// one wave32 per node, 4 floats per lane (128 = 32*4). In-place safe.
// ---------------------------------------------------------------------------
__global__ __launch_bounds__(256) void prep_kernel(const float* __restrict__ X,
                                                   const float* __restrict__ mask,
                                                   float* __restrict__ Y,
                                                   int N, int do_log) {
  int node = (int)((blockIdx.x * blockDim.x + threadIdx.x) >> 5);
  int lane = threadIdx.x & 31;
  if (node >= N) return;
  const float4 v = *(const float4*)(X + (size_t)node * DD + lane * 4);
  float s;
  if (do_log) {
    float ss = v.x * v.x + v.y * v.y + v.z * v.z + v.w * v.w;
    #pragma unroll
    for (int off = 16; off > 0; off >>= 1) ss += __shfl_xor(ss, off, 32);
    float nrm = sqrtf(ss);
    float nc = fminf(fmaxf(nrm, EPSF), 1.0f - EPSF);
    s = atanhf(nc) / fmaxf(nrm, EPSF);
  } else {
    s = 1.0f;
  }
  s *= mask[node];
  float4 o;
  o.x = v.x * s; o.y = v.y * s; o.z = v.z * s; o.w = v.w * s;
  *(float4*)(Y + (size_t)node * DD + lane * 4) = o;
}

// ---------------------------------------------------------------------------
// gemm: Y = (X @ W) * mask, X:[N,128], W:[128,128] row-major.
// One wave per 16x16 output tile. K swept 4 at a time with
// V_WMMA_F32_16X16X4_F32; two independent accumulator chains for ILP.
//
// fp32 WMMA VGPR layouts (ISA 7.12.2):
//   A 16x4 : lane t=lane&15 -> M=t ; half=lane>>4 -> K base 2*half; v2f = K,K+1
//   B 4x16 : lane t -> N=t ; half -> K rows {0,1} / {2,3} (row-striped)
//   C/D    : v8f, VGPR j -> M = j + 8*half, N = t
// ---------------------------------------------------------------------------
__global__ __launch_bounds__(256) void gemm_kernel(const float* __restrict__ X,
                                                   const float* __restrict__ W,
                                                   const float* __restrict__ mask,
                                                   float* __restrict__ Y,
                                                   int N) {
  int wave = (int)((blockIdx.x * blockDim.x + threadIdx.x) >> 5);
  int lane = threadIdx.x & 31;
  int rowTile = wave >> 3;           // D/16 = 8 column tiles
  int colTile = wave & 7;
  int row0 = rowTile * 16;
  if (row0 >= N) return;
  int half = lane >> 4;              // 0 or 1
  int t    = lane & 15;              // M for A, N for B/C/D
  int arow = row0 + t;
  if (arow >= N) arow = N - 1;       // clamp (N is a multiple of 16 anyway)
  const float* Arow  = X + (size_t)arow * DD;
  const float* Bbase = W + colTile * 16 + t;

  v8f c0 = {};
  v8f c1 = {};
  #pragma unroll
  for (int kk = 0; kk < DD; kk += 8) {
    int ka = kk + 2 * half;
    v2f a0 = { Arow[ka], Arow[ka + 1] };
    v2f b0 = { Bbase[(size_t)ka * DD], Bbase[(size_t)(ka + 1) * DD] };
    c0 = __builtin_amdgcn_wmma_f32_16x16x4_f32(false, a0, false, b0,
                                               (short)0, c0, false, false);
    int kb = kk + 4 + 2 * half;
    v2f a1 = { Arow[kb], Arow[kb + 1] };
    v2f b1 = { Bbase[(size_t)kb * DD], Bbase[(size_t)(kb + 1) * DD] };
    c1 = __builtin_amdgcn_wmma_f32_16x16x4_f32(false, a1, false, b1,
                                               (short)0, c1, false, false);
  }

  int col = colTile * 16 + t;
  #pragma unroll
  for (int j = 0; j < 8; ++j) {
    int r = row0 + j + 8 * half;
    if (r < N) {
      Y[(size_t)r * DD + col] = (c0[j] + c1[j]) * mask[r];
    }
  }
}

// ---------------------------------------------------------------------------
// agg: out[n,:] = relu(exp_map_zero((sum_k w[n,k]*msg[adj[n,k],:]) * m)) * m
// one wave32 per node; lane k holds adj/weight element k, broadcast by __shfl.
// Gathers are 512B rows -> float4 per lane, L2-resident.
// ---------------------------------------------------------------------------
__global__ __launch_bounds__(256) void agg_kernel(const float* __restrict__ msg,
                                                  const int* __restrict__ adj,
                                                  const float* __restrict__ wgt,
                                                  const float* __restrict__ mask,
                                                  float* __restrict__ out,
                                                  int N) {
  int node = (int)((blockIdx.x * blockDim.x + threadIdx.x) >> 5);
  int lane = threadIdx.x & 31;
  if (node >= N) return;

  int   myidx = adj[(size_t)node * KNB + lane];
  float myw   = wgt[(size_t)node * KNB + lane];

  float4 acc; acc.x = acc.y = acc.z = acc.w = 0.0f;
  #pragma unroll 4
  for (int k = 0; k < KNB; ++k) {
    int   idx = __shfl(myidx, k, 32);
    float wk  = __shfl(myw, k, 32);
    if (k + 8 < KNB) {
      int pidx = __shfl(myidx, k + 8, 32);
      __builtin_prefetch(msg + (size_t)pidx * DD + lane * 4, 0, 3);
    }
    const float4 mr = *(const float4*)(msg + (size_t)idx * DD + lane * 4);
    acc.x += wk * mr.x;
    acc.y += wk * mr.y;
    acc.z += wk * mr.z;
    acc.w += wk * mr.w;
  }

  float m = mask[node];
  acc.x *= m; acc.y *= m; acc.z *= m; acc.w *= m;

  float ss = acc.x * acc.x + acc.y * acc.y + acc.z * acc.z + acc.w * acc.w;
  #pragma unroll
  for (int off = 16; off > 0; off >>= 1) ss += __shfl_xor(ss, off, 32);
  float nrm = sqrtf(ss);
  float nc = fminf(fmaxf(nrm, EPSF), MAXTANH);
  float s = tanhf(nc) / fmaxf(nrm, EPSF);

  float4 o;
  o.x = fmaxf(acc.x * s * m, 0.0f) * m;
  o.y = fmaxf(acc.y * s * m, 0.0f) * m;
  o.z = fmaxf(acc.z * s * m, 0.0f) * m;
  o.w = fmaxf(acc.w * s * m, 0.0f) * m;
  *(float4*)(out + (size_t)node * DD + lane * 4) = o;
}

// ---------------------------------------------------------------------------
// host launcher
// inputs: 0=node_repr[N,128] f32, 1=adj_mat[N,32] i32, 2=weight[N,32] f32,
//         3=mask[N,1] f32, 4=msg_weights[2,128,128] f32
// ---------------------------------------------------------------------------
extern "C" void kernel_launch(void* const* d_in, const int* in_sizes, int n_in,
                              void* d_out, int out_size, void* d_ws, size_t ws_size,
                              hipStream_t stream) {
  (void)n_in; (void)out_size; (void)ws_size;
  const float* node_repr = (const float*)d_in[0];
  const int*   adj       = (const int*)d_in[1];
  const float* wgt       = (const float*)d_in[2];
  const float* mask      = (const float*)d_in[3];
  const float* msgW      = (const float*)d_in[4];
  float*       out       = (float*)d_out;

  const int N = in_sizes[3];  // mask has N elements

  float* ws0 = (float*)d_ws;                 // x / prepped x   [N*128]
  float* ws1 = ws0 + (size_t)N * DD;         // msg             [N*128]

  const int nodeBlocks = (N + 7) / 8;                 // 8 waves/block, 1 node/wave
  const int gemmWaves  = ((N + 15) / 16) * (DD / 16); // one wave per 16x16 tile
  const int gemmBlocks = (gemmWaves + 7) / 8;

  // ----- layer 0 -----
  prep_kernel<<<nodeBlocks, 256, 0, stream>>>(node_repr, mask, ws0, N, 0);
  gemm_kernel<<<gemmBlocks, 256, 0, stream>>>(ws0, msgW, mask, ws1, N);
  agg_kernel<<<nodeBlocks, 256, 0, stream>>>(ws1, adj, wgt, mask, ws0, N);

  // ----- layer 1 -----
  prep_kernel<<<nodeBlocks, 256, 0, stream>>>(ws0, mask, ws0, N, 1);
  gemm_kernel<<<gemmBlocks, 256, 0, stream>>>(ws0, msgW + DD * DD, mask, ws1, N);
  agg_kernel<<<nodeBlocks, 256, 0, stream>>>(ws1, adj, wgt, mask, out, N);
}